// P20RotaryStateOutputRuntimeSequenceMixer_9543417332017
// MI455X (gfx1250) — compile-verified
//
#include <hip/hip_runtime.h>

// ---------------------------------------------------------------------------
// Problem constants (from reference): B=16, S=2048, D=1024
// ---------------------------------------------------------------------------
#define Bb   16
#define Ss   2048
#define Dd   1024
#define Hh   512            // D/2
#define Pp   3584           // 3*D + D/2
#define Mm   (Bb * Ss)      // 32768 rows of the projection GEMM
#define NBLK_SCAN 64        // workgroups cooperating on the scan (16 cols each)

typedef __attribute__((ext_vector_type(2))) float v2f;
typedef __attribute__((ext_vector_type(8))) float v8f;

__device__ __forceinline__ float sigmoid_f(float v) {
    return 1.0f / (1.0f + __expf(-v));
}
__device__ __forceinline__ float tanh_f(float v) {
    // tanh(v) = 1 - 2/(exp(2v)+1); correct limits at +-inf
    return 1.0f - 2.0f / (__expf(2.0f * v) + 1.0f);
}

// ---------------------------------------------------------------------------
// Kernel 0: zero the ping-pong state buffers and the per-step barrier counters
// (must run every launch: harness does not re-poison between graph replays)
// ---------------------------------------------------------------------------
__global__ void rsq_init(float* __restrict__ st0, float* __restrict__ st1,
                         int* __restrict__ cnt) {
    int i = blockIdx.x * blockDim.x + threadIdx.x;
    if (i < Bb * Dd) { st0[i] = 0.0f; st1[i] = 0.0f; }
    if (i < Ss) cnt[i] = 0;
}

// ---------------------------------------------------------------------------
// Kernel 1: projection GEMM  proj = x @ Wp.T + bp, fused activation epilogue.
// Block = 256 threads = 8 waves arranged 2(M) x 4(N); each wave owns a 64x64
// output strip: 4x4 accumulator tiles, 16 v_wmma_f32_16x16x4_f32 per K-step
// fed by 8 x 8B loads (2 WMMAs per load), double-buffered in registers.
// Loads are expressed as sgpr_base + 32-bit index so the compiler uses the
// GVS addressing form (saddr + voffset) instead of 64-bit VALU address adds.
// Activation outputs (written once, read once, ~537MB) are stored with
// non-temporal hints so they stream to HBM without evicting hot L2 data.
// Grid = (Pp/256, Mm/128).
// ---------------------------------------------------------------------------
__global__ __launch_bounds__(256) void rsq_proj_gemm(
    const float* __restrict__ x, const float* __restrict__ Wp,
    const float* __restrict__ bp,
    float* __restrict__ ubuf, float* __restrict__ cosbuf,
    float* __restrict__ sinbuf, float* __restrict__ candbuf,
    float* __restrict__ ogbuf)
{
    const int lane  = threadIdx.x & 31;
    const int wave  = threadIdx.x >> 5;
    const int wm    = wave & 1;             // 2 wave-rows (M)
    const int wn    = wave >> 1;            // 4 wave-cols (N)
    const int lrow  = lane & 15;            // M (for A) / N (for B) within tile
    const int khalf = (lane >> 4) << 1;     // 0 or 2: K sub-offset per lane half

    const int r0 = blockIdx.y * 128 + wm * 64;   // first output row of wave
    const int n0 = blockIdx.x * 256 + wn * 64;   // first output col of wave

    // 32-bit element indices (max ~33.5M floats < 2^31 bytes): GVS addressing
    int xi[4], wi[4];
    #pragma unroll
    for (int i = 0; i < 4; ++i) {
        xi[i] = (r0 + i * 16 + lrow) * Dd + khalf;
        wi[i] = (n0 + i * 16 + lrow) * Dd + khalf;
    }

    v8f acc[4][4];
    #pragma unroll
    for (int mt = 0; mt < 4; ++mt)
        #pragma unroll
        for (int nt = 0; nt < 4; ++nt) acc[mt][nt] = v8f{};

    v2f a0[4], b0[4], a1[4], b1[4];

    #define LOAD_SET(A, B, kk)                                         \
        do {                                                           \
            _Pragma("unroll")                                          \
            for (int i = 0; i < 4; ++i) {                              \
                (A)[i] = *(const v2f*)(x  + (xi[i] + (kk)));           \
                (B)[i] = *(const v2f*)(Wp + (wi[i] + (kk)));           \
            }                                                          \
        } while (0)

    #define WMMA_SET(A, B)                                             \
        do {                                                           \
            _Pragma("unroll")                                          \
            for (int mt = 0; mt < 4; ++mt)                             \
                _Pragma("unroll")                                      \
                for (int nt = 0; nt < 4; ++nt)                         \
                    acc[mt][nt] = __builtin_amdgcn_wmma_f32_16x16x4_f32( \
                        false, (A)[mt], false, (B)[nt], (short)0,      \
                        acc[mt][nt], false, false);                    \
        } while (0)

    LOAD_SET(a0, b0, 0);
    #pragma unroll 1
    for (int k = 0; k < Dd; k += 8) {
        LOAD_SET(a1, b1, k + 4);            // in flight under set-0 WMMAs
        WMMA_SET(a0, b0);
        const int kn = (k + 8 < Dd) ? (k + 8) : k;   // clamped tail (unused)
        LOAD_SET(a0, b0, kn);               // in flight under set-1 WMMAs
        WMMA_SET(a1, b1);
    }
    #undef LOAD_SET
    #undef WMMA_SET

    // Epilogue: C/D layout — element i of lane L -> (m = i + 8*(L>>4), n = L&15)
    const int mbase = (lane >> 4) << 3;
    #pragma unroll
    for (int mt = 0; mt < 4; ++mt) {
        #pragma unroll
        for (int nt = 0; nt < 4; ++nt) {
            const int c = n0 + nt * 16 + (lane & 15);
            const float bias = bp[c];
            #pragma unroll
            for (int i = 0; i < 8; ++i) {
                const int r = r0 + mt * 16 + mbase + i;
                const float v = acc[mt][nt][i] + bias;
                if (c < Dd) {                       // update gate: sigmoid
                    __builtin_nontemporal_store(sigmoid_f(v), ubuf + r * Dd + c);
                } else if (c < Dd + Hh) {           // rotation angle: cos/sin
                    const int j = c - Dd;
                    float sv, cv;
                    __sincosf(v, &sv, &cv);
                    __builtin_nontemporal_store(cv, cosbuf + r * Hh + j);
                    __builtin_nontemporal_store(sv, sinbuf + r * Hh + j);
                } else if (c < 2 * Dd + Hh) {       // candidate: tanh
                    __builtin_nontemporal_store(tanh_f(v),
                                                candbuf + r * Dd + (c - (Dd + Hh)));
                } else {                            // output gate: sigmoid
                    __builtin_nontemporal_store(sigmoid_f(v),
                                                ogbuf + r * Dd + (c - (2 * Dd + Hh)));
                }
            }
        }
    }
}

// ---------------------------------------------------------------------------
// Kernel 2: sequential rotary scan. 64 blocks x 256 threads; block owns 16
// state columns. Each wave holds its 16x128 Wt slice in registers; per step
// it issues all 32 state-fragment loads plus the 5 activation loads (37 in
// flight, LOADcnt max 63), then streams 32 v_wmma_f32_16x16x4_f32 while they
// drain. Partials are tree-reduced in LDS in fixed order (deterministic).
// Activation streams are loaded non-temporally (read-once) so Wt fragments
// and the state ping-pong buffers stay L2-resident across all 2048 steps.
// Per-step device-wide sync via release/acquire atomic counters.
// ---------------------------------------------------------------------------
__global__ __launch_bounds__(256) void rsq_scan(
    const float* __restrict__ Wt, const float* __restrict__ bt,
    const float* __restrict__ ubuf, const float* __restrict__ cosbuf,
    const float* __restrict__ sinbuf, const float* __restrict__ candbuf,
    const float* __restrict__ ogbuf,
    float* __restrict__ out, float* __restrict__ fstate,
    float* __restrict__ st0, float* __restrict__ st1,
    int* __restrict__ cnt)
{
    __shared__ float part[8][16][16];   // per-wave partial tiles
    __shared__ float pstile[16][17];    // biased ps tile (padded rows)

    const int lane  = threadIdx.x & 31;
    const int wave  = threadIdx.x >> 5;
    const int lrow  = lane & 15;
    const int khalf = (lane >> 4) << 1;
    const int col0  = blockIdx.x * 16;          // 16 owned state columns

    // Preload B fragments: Wt rows (col0+lrow), K slice [wave*128, wave*128+128)
    v2f bfrag[32];
    {
        const int wbase = (col0 + lrow) * Dd + wave * 128 + khalf;
        #pragma unroll
        for (int kk = 0; kk < 32; ++kk) bfrag[kk] = *(const v2f*)(Wt + (wbase + kk * 4));
    }

    // Epilogue role of this thread: (m = batch row, n = local column)
    const int m = threadIdx.x >> 4;      // 0..15
    const int n = threadIdx.x & 15;      // 0..15
    const int c = col0 + n;              // global state column
    const float btv = bt[c];
    const int actRow = m * Ss;           // row base into (B,S,*) buffers

    const int abase = lrow * Dd + wave * 128 + khalf;   // A-frag offset

    for (int t = 0; t < Ss; ++t) {
        const float* __restrict__ Scur = (t & 1) ? st1 : st0;
        float* __restrict__       Snxt = (t & 1) ? st0 : st1;

        // ---- issue all A-fragment loads + activation loads, then WMMAs ----
        v2f afrag[32];
        #pragma unroll
        for (int kk = 0; kk < 32; ++kk)
            afrag[kk] = *(const v2f*)(Scur + (abase + kk * 4));

        const int aidx = (actRow + t) * Dd + c;
        const int ridx = (actRow + t) * Hh + (c >> 1);
        const float cosv = __builtin_nontemporal_load(cosbuf + ridx);
        const float sinv = __builtin_nontemporal_load(sinbuf + ridx);
        const float uv   = __builtin_nontemporal_load(ubuf + aidx);
        const float cv   = __builtin_nontemporal_load(candbuf + aidx);
        const float ov   = __builtin_nontemporal_load(ogbuf + aidx);

        v8f acc = {};
        #pragma unroll
        for (int kk = 0; kk < 32; ++kk) {
            acc = __builtin_amdgcn_wmma_f32_16x16x4_f32(false, afrag[kk],
                                                        false, bfrag[kk],
                                                        (short)0, acc, false, false);
        }
        {
            const int mb = (lane >> 4) << 3;
            #pragma unroll
            for (int i = 0; i < 8; ++i) part[wave][mb + i][lane & 15] = acc[i];
        }
        __syncthreads();

        // ---- deterministic tree reduction + bias ----
        float ps = btv;
        #pragma unroll
        for (int w = 0; w < 8; ++w) ps += part[w][m][n];
        pstile[m][n] = ps;
        __syncthreads();
        const float psp = pstile[m][n ^ 1];

        // ---- rotation + gating ----
        const float ts = (n & 1) ? (psp * sinv + ps * cosv)
                                 : (ps * cosv - psp * sinv);
        const float ns = uv * ts + (1.0f - uv) * cv;

        Snxt[m * Dd + c] = ns;
        __builtin_nontemporal_store(ov * ns, out + aidx);
        if (t == Ss - 1) fstate[m * Dd + c] = ns;

        // ---- device-wide step barrier (release/acquire) ----
        __threadfence();
        __syncthreads();
        if (threadIdx.x == 0) {
            __hip_atomic_fetch_add(cnt + t, 1, __ATOMIC_RELEASE,
                                   __HIP_MEMORY_SCOPE_AGENT);
            while (__hip_atomic_load(cnt + t, __ATOMIC_ACQUIRE,
                                     __HIP_MEMORY_SCOPE_AGENT) < NBLK_SCAN) {
                __builtin_amdgcn_s_sleep(1);
            }
        }
        __syncthreads();
    }
}

// ---------------------------------------------------------------------------
// Host launcher. Inputs: x, Wp, bp, Wt, bt (fp32). Output: outs (B,S,D) then
// final_state (B,D). Workspace layout (floats):
//   u[ M*D ] | cos[ M*H ] | sin[ M*H ] | cand[ M*D ] | og[ M*D ]
//   | st0[B*D] | st1[B*D] | counters[S] (int)           (~537 MB total)
// ---------------------------------------------------------------------------
extern "C" void kernel_launch(void* const* d_in, const int* in_sizes, int n_in,
                              void* d_out, int out_size, void* d_ws, size_t ws_size,
                              hipStream_t stream) {
    const float* x  = (const float*)d_in[0];
    const float* Wp = (const float*)d_in[1];
    const float* bp = (const float*)d_in[2];
    const float* Wt = (const float*)d_in[3];
    const float* bt = (const float*)d_in[4];

    float* out    = (float*)d_out;                       // (B,S,D)
    float* fstate = out + (size_t)Mm * Dd;               // (B,D)

    float* ws     = (float*)d_ws;
    float* ubuf   = ws;
    float* cosbuf = ubuf   + (size_t)Mm * Dd;
    float* sinbuf = cosbuf + (size_t)Mm * Hh;
    float* candbuf= sinbuf + (size_t)Mm * Hh;
    float* ogbuf  = candbuf+ (size_t)Mm * Dd;
    float* st0    = ogbuf  + (size_t)Mm * Dd;
    float* st1    = st0 + Bb * Dd;
    int*   cnt    = (int*)(st1 + Bb * Dd);

    rsq_init<<<64, 256, 0, stream>>>(st0, st1, cnt);

    dim3 gGemm(Pp / 256, Mm / 128);
    rsq_proj_gemm<<<gGemm, 256, 0, stream>>>(x, Wp, bp,
                                             ubuf, cosbuf, sinbuf, candbuf, ogbuf);

    rsq_scan<<<NBLK_SCAN, 256, 0, stream>>>(Wt, bt, ubuf, cosbuf, sinbuf,
                                            candbuf, ogbuf, out, fstate,
                                            st0, st1, cnt);
}